// KeepTopK_66571993088216
// MI455X (gfx1250) — compile-verified
//
#include <hip/hip_runtime.h>
#include <stdint.h>

// KeepTopK: out[i,j] = x[i,j] if x[i,j] is among row i's top-8, else BETA.
// N=262144 rows, E=256 cols, fp32. Memory-bound (512 MB @ 23.3 TB/s ~ 22us).
// CDNA5 path: double-buffered GLOBAL_LOAD_ASYNC_TO_LDS_B128 (ASYNCcnt) DMA
// pipeline with NT cache hints, wave32 ds_swizzle xor-reductions, one wave/row.

#define E_DIM 256
#define TOPK 8
#define BETA_F 1000000.0f
#define ROWS_PER_WAVE 8
#define WAVES_PER_BLOCK 8
#define BLOCK_THREADS (WAVES_PER_BLOCK * 32)

typedef float v4f __attribute__((ext_vector_type(4)));

// Wave32 max-reduce using group-of-32 xor swizzles (immediate-offset
// ds_swizzle_b32: offset = (xor_mask << 10) | and_mask(0x1f); offset[15]=0).
__device__ __forceinline__ float wave_max32(float v) {
  float t;
  t = __int_as_float(__builtin_amdgcn_ds_swizzle(__float_as_int(v), 0x401F)); // SWAPX16
  v = fmaxf(v, t);
  t = __int_as_float(__builtin_amdgcn_ds_swizzle(__float_as_int(v), 0x201F)); // SWAPX8
  v = fmaxf(v, t);
  t = __int_as_float(__builtin_amdgcn_ds_swizzle(__float_as_int(v), 0x101F)); // SWAPX4
  v = fmaxf(v, t);
  t = __int_as_float(__builtin_amdgcn_ds_swizzle(__float_as_int(v), 0x081F)); // SWAPX2
  v = fmaxf(v, t);
  t = __int_as_float(__builtin_amdgcn_ds_swizzle(__float_as_int(v), 0x041F)); // SWAPX1
  v = fmaxf(v, t);
  return v;
}

__global__ __launch_bounds__(BLOCK_THREADS)
void keep_topk_kernel(const float* __restrict__ x, float* __restrict__ out,
                      int nrows) {
  __shared__ float smem[WAVES_PER_BLOCK * 2 * E_DIM];  // 16 KB: 2 row-buffers/wave
  const int lane = (int)(threadIdx.x & 31);
  const int wave = (int)(threadIdx.x >> 5);
  const long long waveId = (long long)blockIdx.x * WAVES_PER_BLOCK + wave;
  const long long row0 = waveId * ROWS_PER_WAVE;

  float* wbuf = &smem[wave * (2 * E_DIM)];
  // Low 32 bits of the flat address of a __shared__ object == LDS byte offset
  // (flat LDS aperture keeps the offset in addr[31:0]).
  const uint32_t lds_lane_base = (uint32_t)(uintptr_t)wbuf + (uint32_t)(lane * 16);

  // Async-DMA one full row (1024B) into LDS double-buffer `buf`.
  // Two GLOBAL_LOAD_ASYNC_TO_LDS_B128 (32 lanes x 16B = 512B each); the 24-bit
  // instruction offset is added to BOTH the LDS and global addresses, so a
  // single +512 offset covers the second half-row. TH_LOAD_NT: one-pass
  // 512 MB stream vs 192 MB L2 -> do not claim L2 residency.
  auto issue_row = [&](int rr, int buf) {
    const long long row = row0 + rr;
    if (row < nrows) {
      const uint64_t g = (uint64_t)(uintptr_t)(x + row * (long long)E_DIM) +
                         (uint64_t)(uint32_t)(lane * 16);
      const uint32_t l = lds_lane_base + (uint32_t)(buf * (E_DIM * 4));
      asm volatile(
          "global_load_async_to_lds_b128 %0, %1, off th:TH_LOAD_NT\n\t"
          "global_load_async_to_lds_b128 %0, %1, off offset:512 th:TH_LOAD_NT"
          :: "v"(l), "v"(g)
          : "memory");
    }
  };

  issue_row(0, 0);

  for (int r = 0; r < ROWS_PER_WAVE; ++r) {
    const int cur = r & 1;
    if (r + 1 < ROWS_PER_WAVE) {
      issue_row(r + 1, cur ^ 1);  // prefetch next row into the other buffer
      // ASYNCcnt decrements in order: <=2 outstanding means row r's two
      // B128 transfers have landed in LDS.
      asm volatile("s_wait_asynccnt 0x2" ::: "memory");
    } else {
      asm volatile("s_wait_asynccnt 0x0" ::: "memory");
    }

    const long long row = row0 + r;
    if (row >= nrows) continue;

    // Read this wave's row back from LDS: 8 floats per lane, two runs of 4 floats 128 apart.
    const v4f* bp = (const v4f*)(wbuf + cur * E_DIM);
    const v4f a = bp[lane];       // elements [4*lane .. 4*lane+3]
    const v4f b = bp[lane + 32];  // elements [128+4*lane .. 128+4*lane+3]

    float v0 = a.x, v1 = a.y, v2 = a.z, v3 = a.w;
    float v4 = b.x, v5 = b.y, v6 = b.z, v7 = b.w;
    float m = fmaxf(fmaxf(fmaxf(v0, v1), fmaxf(v2, v3)),
                    fmaxf(fmaxf(v4, v5), fmaxf(v6, v7)));

    const float NEGINF = -__builtin_inff();
    float thresh = NEGINF;
#pragma unroll
    for (int k = 0; k < TOPK; ++k) {
      const float g = wave_max32(m);  // k-th largest value of the row
      thresh = g;
      if (k < TOPK - 1) {
        // Remove the extracted value from this lane's working set.
        v0 = (v0 == g) ? NEGINF : v0;
        v1 = (v1 == g) ? NEGINF : v1;
        v2 = (v2 == g) ? NEGINF : v2;
        v3 = (v3 == g) ? NEGINF : v3;
        v4 = (v4 == g) ? NEGINF : v4;
        v5 = (v5 == g) ? NEGINF : v5;
        v6 = (v6 == g) ? NEGINF : v6;
        v7 = (v7 == g) ? NEGINF : v7;
        m = fmaxf(fmaxf(fmaxf(v0, v1), fmaxf(v2, v3)),
                  fmaxf(fmaxf(v4, v5), fmaxf(v6, v7)));
      }
    }

    // thresh == 8th-largest value of the row: keep >= thresh, else BETA.
    v4f oa, ob;
    oa.x = (a.x >= thresh) ? a.x : BETA_F;
    oa.y = (a.y >= thresh) ? a.y : BETA_F;
    oa.z = (a.z >= thresh) ? a.z : BETA_F;
    oa.w = (a.w >= thresh) ? a.w : BETA_F;
    ob.x = (b.x >= thresh) ? b.x : BETA_F;
    ob.y = (b.y >= thresh) ? b.y : BETA_F;
    ob.z = (b.z >= thresh) ? b.z : BETA_F;
    ob.w = (b.w >= thresh) ? b.w : BETA_F;

    // Non-temporal full-cacheline stores: pure write stream, no reuse.
    v4f* op = (v4f*)(out + row * (long long)E_DIM);
    __builtin_nontemporal_store(oa, op + lane);
    __builtin_nontemporal_store(ob, op + lane + 32);
  }
}

extern "C" void kernel_launch(void* const* d_in, const int* in_sizes, int n_in,
                              void* d_out, int out_size, void* d_ws,
                              size_t ws_size, hipStream_t stream) {
  const float* x = (const float*)d_in[0];
  float* out = (float*)d_out;
  const int total = in_sizes[0];          // N * E
  const int nrows = total / E_DIM;        // 262144
  const int rows_per_block = ROWS_PER_WAVE * WAVES_PER_BLOCK;  // 64
  const int blocks = (nrows + rows_per_block - 1) / rows_per_block;  // 4096
  keep_topk_kernel<<<blocks, BLOCK_THREADS, 0, stream>>>(x, out, nrows);
}